// GCNNet_78400333021315
// MI455X (gfx1250) — compile-verified
//
#include <hip/hip_runtime.h>
#include <math.h>

#define N_NODES 100000
#define N_EDGES 500000
#define NB 512

typedef float v2f __attribute__((ext_vector_type(2)));
typedef float v8f __attribute__((ext_vector_type(8)));

// ---------------------------------------------------------------- utilities
__global__ void fill_kernel(float* __restrict__ p, float v, int n) {
    int i = blockIdx.x * blockDim.x + threadIdx.x;
    if (i < n) p[i] = v;
}

__global__ void deg_count(const int* __restrict__ dst, float* __restrict__ deg) {
    int e = blockIdx.x * blockDim.x + threadIdx.x;
    if (e < N_EDGES) atomicAdd(&deg[dst[e]], 1.0f);
}

__global__ void make_dinv(float* __restrict__ d) {
    int i = blockIdx.x * blockDim.x + threadIdx.x;
    if (i < N_NODES) d[i] = rsqrtf(d[i]);
}

// Pack weight W[K,N] (row-major) into zero-padded pair-packed Bp:
// Bp element pair p=(k>>1), col n:  Bp[(p*Np + n)*2 + (k&1)] = W[k,n]
// so a WMMA lane's two B values (rows ka,ka+1 of column n) are one b64 load.
__global__ void pack_B(const float* __restrict__ W, float* __restrict__ Bp,
                       int K, int N, int Kp, int Np) {
    int idx = blockIdx.x * blockDim.x + threadIdx.x;
    int tot = Kp * Np;
    if (idx >= tot) return;
    int k = idx / Np;
    int n = idx % Np;
    float v = (k < K && n < N) ? W[(size_t)k * N + n] : 0.0f;
    Bp[((size_t)(k >> 1) * Np + n) * 2 + (k & 1)] = v;
}

// msg[i, 0:Kp] = x[i, 0:F] * dinv[i]^2, zero-padded columns F..Kp
__global__ void self_msg(const float* __restrict__ x, const float* __restrict__ dinv,
                         float* __restrict__ msg, int F, int Kp) {
    long long idx = (long long)blockIdx.x * blockDim.x + threadIdx.x;
    long long tot = (long long)N_NODES * Kp;
    if (idx >= tot) return;
    int i = (int)(idx / Kp);
    int f = (int)(idx % Kp);
    float di = dinv[i];
    msg[idx] = (f < F) ? x[(size_t)i * F + f] * di * di : 0.0f;
}

// one block per edge, lanes over features: msg[dst] += x[src] * dinv[src]*dinv[dst]
__global__ void edge_agg(const float* __restrict__ x, const float* __restrict__ dinv,
                         const int* __restrict__ src, const int* __restrict__ dst,
                         float* __restrict__ msg, int F, int Kp) {
    int e = blockIdx.x;
    int f = threadIdx.x;
    if (f >= F) return;
    int s = src[e], d = dst[e];
    float c = dinv[s] * dinv[d];
    atomicAdd(&msg[(long long)d * Kp + f], x[(long long)s * F + f] * c);
}

// segment max via uint-bit atomicMax (valid: h >= 0 after relu, g zero-initialized)
__global__ void seg_max(const float* __restrict__ h, const int* __restrict__ batch,
                        float* __restrict__ g) {
    long long idx = (long long)blockIdx.x * blockDim.x + threadIdx.x;
    long long tot = (long long)N_NODES * 312;
    if (idx >= tot) return;
    int i = (int)(idx / 312);
    int f = (int)(idx % 312);
    unsigned v = __float_as_uint(h[idx]);
    atomicMax((unsigned*)&g[(long long)batch[i] * 312 + f], v);
}

// Conv1d: target [B,750,19] (NCH), Kxt [32,750,8] (OIH), VALID -> out [B, 32*12]
__global__ void conv1d_k(const float* __restrict__ T, const float* __restrict__ Kw,
                         const float* __restrict__ bias, float* __restrict__ out) {
    int idx = blockIdx.x * blockDim.x + threadIdx.x;
    if (idx >= NB * 384) return;
    int b = idx / 384;
    int r = idx % 384;
    int o = r / 12;
    int j = r % 12;
    const float* tb = T + (size_t)b * 750 * 19 + j;
    const float* kb = Kw + (size_t)o * 750 * 8;
    float acc = bias[o];
    for (int i = 0; i < 750; ++i) {
#pragma unroll
        for (int kk = 0; kk < 8; ++kk)
            acc = fmaf(tb[i * 19 + kk], kb[i * 8 + kk], acc);
    }
    out[idx] = acc;
}

// ------------------------------------------------------------- WMMA GEMM f32
// C[M,Nreal] = relu?( A[M,K] @ W + bias ), A row-major with stride lda (even,
// K multiple of 4), W pre-packed as Bp (see pack_B, Np multiple of 16).
// One wave -> one 16x16 C tile via V_WMMA_F32_16X16X4_F32. Inner loop is
// branchless: one b64 A load + one coalesced b64 B load + one WMMA per K-step.
// blockDim = (32, 8): 8 waves cover 8 consecutive N tiles.
__global__ void wmma_gemm(const float* __restrict__ A, const float* __restrict__ Bp,
                          const float* __restrict__ bias, float* __restrict__ C,
                          int lda, int Np, int K, int Nreal, int ldc, int relu) {
    int lane = threadIdx.x;          // 0..31
    int hf   = lane >> 4;            // 0: K pair {0,1}; 1: K pair {2,3}
    int l    = lane & 15;
    int n0   = (blockIdx.x * blockDim.y + threadIdx.y) * 16;
    int m0   = blockIdx.y * 16;
    if (n0 >= Np) return;            // whole idle waves exit

    int nc = n0 + l;
    const v2f* A2 = (const v2f*)(A + (size_t)(m0 + l) * lda) + hf;   // pair idx 2s+hf
    const v2f* B2 = (const v2f*)Bp + (size_t)hf * Np + nc;           // pair idx (2s+hf)*Np+nc

    v8f c = {};
    int steps = K >> 2;
#pragma unroll 4
    for (int s = 0; s < steps; ++s) {
        v2f a = A2[(size_t)2 * s];
        v2f b = B2[(size_t)2 * s * Np];
        c = __builtin_amdgcn_wmma_f32_16x16x4_f32(false, a, false, b,
                                                  (short)0, c, false, false);
    }
    if (nc < Nreal) {
        float bv = bias[nc];
#pragma unroll
        for (int v = 0; v < 8; ++v) {           // C/D: VGPR v -> row v + 8*hf
            float val = c[v] + bv;
            if (relu) val = fmaxf(val, 0.0f);
            C[(size_t)(m0 + v + hf * 8) * ldc + nc] = val;
        }
    }
}

// ------------------------------------------------------------------ launcher
extern "C" void kernel_launch(void* const* d_in, const int* in_sizes, int n_in,
                              void* d_out, int out_size, void* d_ws, size_t ws_size,
                              hipStream_t stream) {
    const float* x    = (const float*)d_in[0];
    const int*   ei   = (const int*)d_in[1];
    const int*   bat  = (const int*)d_in[2];
    const float* tgt  = (const float*)d_in[3];
    const float* W1   = (const float*)d_in[4];  const float* b1   = (const float*)d_in[5];
    const float* W2   = (const float*)d_in[6];  const float* b2   = (const float*)d_in[7];
    const float* W3   = (const float*)d_in[8];  const float* b3   = (const float*)d_in[9];
    const float* Wg1  = (const float*)d_in[10]; const float* bg1  = (const float*)d_in[11];
    const float* Wg2  = (const float*)d_in[12]; const float* bg2  = (const float*)d_in[13];
    const float* Kxt  = (const float*)d_in[14]; const float* bxt  = (const float*)d_in[15];
    const float* Wxt  = (const float*)d_in[16]; const float* bxt2 = (const float*)d_in[17];
    const float* Wf1  = (const float*)d_in[18]; const float* bf1  = (const float*)d_in[19];
    const float* Wf2  = (const float*)d_in[20]; const float* bf2  = (const float*)d_in[21];
    const float* Wo   = (const float*)d_in[22]; const float* bo   = (const float*)d_in[23];

    const int* src = ei;
    const int* dst = ei + N_EDGES;

    // workspace carve-out (floats, 256B-aligned chunks); ~253 MB + ~6 MB packed weights
    float* p = (float*)d_ws;
    auto take = [&](size_t n) { float* r = p; p += (n + 63) & ~(size_t)63; return r; };
    float* DINV = take(N_NODES);
    float* MSG  = take((size_t)N_NODES * 156);   // aggregated input features (max Kp=156)
    float* HA   = take((size_t)N_NODES * 312);   // h1, then h3
    float* HB   = take((size_t)N_NODES * 156);   // h2
    float* G    = take((size_t)NB * 312);
    float* GH   = take((size_t)NB * 1024);
    float* XC   = take((size_t)NB * 256);
    float* CONV = take((size_t)NB * 384);
    float* F1   = take((size_t)NB * 1024);
    float* F2   = take((size_t)NB * 512);
    // packed weights (Kp x Np, pair-packed)
    float* W1p  = take((size_t)80 * 80);
    float* W2p  = take((size_t)80 * 160);
    float* W3p  = take((size_t)156 * 320);
    float* Wg1p = take((size_t)312 * 1024);
    float* Wg2p = take((size_t)1024 * 128);
    float* Wxtp = take((size_t)384 * 128);
    float* Wf1p = take((size_t)256 * 1024);
    float* Wf2p = take((size_t)1024 * 512);
    float* Wop  = take((size_t)512 * 16);

    auto pack = [&](const float* W, float* Bp, int K, int N, int Kp, int Np) {
        int tot = Kp * Np;
        pack_B<<<(tot + 255) / 256, 256, 0, stream>>>(W, Bp, K, N, Kp, Np);
    };
    pack(W1,  W1p,  78,   78,  80,   80);
    pack(W2,  W2p,  78,  156,  80,  160);
    pack(W3,  W3p, 156,  312, 156,  320);
    pack(Wg1, Wg1p, 312, 1024, 312, 1024);
    pack(Wg2, Wg2p, 1024, 128, 1024, 128);
    pack(Wxt, Wxtp, 384,  128, 384,  128);
    pack(Wf1, Wf1p, 256, 1024, 256, 1024);
    pack(Wf2, Wf2p, 1024, 512, 1024, 512);
    pack(Wo,  Wop,  512,    1, 512,   16);

    // ---- degrees (shared by all three layers): deg = 1 + indegree; dinv = deg^-1/2
    fill_kernel<<<(N_NODES + 255) / 256, 256, 0, stream>>>(DINV, 1.0f, N_NODES);
    deg_count<<<(N_EDGES + 255) / 256, 256, 0, stream>>>(dst, DINV);
    make_dinv<<<(N_NODES + 255) / 256, 256, 0, stream>>>(DINV);

    auto gemm = [&](const float* A, const float* Bp, const float* bb, float* C,
                    int M, int lda, int K, int Np, int Nreal, int ldc, int relu) {
        dim3 blk(32, 8);
        dim3 grd((unsigned)((Np / 16 + 7) / 8), (unsigned)(M / 16));
        wmma_gemm<<<grd, blk, 0, stream>>>(A, Bp, bb, C, lda, Np, K, Nreal, ldc, relu);
    };

    // GCN layer: aggregate first (linearity), then one WMMA GEMM + bias + relu
    auto layer = [&](const float* in, int Fin, int Kp, const float* Bp, const float* bb,
                     int Fout, int Np, float* out) {
        long long tot = (long long)N_NODES * Kp;
        self_msg<<<(unsigned)((tot + 255) / 256), 256, 0, stream>>>(in, DINV, MSG, Fin, Kp);
        int bt = ((Fin + 31) / 32) * 32;
        edge_agg<<<N_EDGES, bt, 0, stream>>>(in, DINV, src, dst, MSG, Fin, Kp);
        gemm(MSG, Bp, bb, out, N_NODES, Kp, Kp, Np, Fout, Fout, 1);
    };

    layer(x,  78,  80,  W1p, b1, 78,  80,  HA);
    layer(HA, 78,  80,  W2p, b2, 156, 160, HB);
    layer(HB, 156, 156, W3p, b3, 312, 320, HA);

    // global max pool over batch segments (zero G every call: atomicMax accumulates)
    fill_kernel<<<(NB * 312 + 255) / 256, 256, 0, stream>>>(G, 0.0f, NB * 312);
    {
        long long tot = (long long)N_NODES * 312;
        seg_max<<<(unsigned)((tot + 255) / 256), 256, 0, stream>>>(HA, bat, G);
    }

    // graph head
    gemm(G,  Wg1p, bg1, GH, NB, 312, 312, 1024, 1024, 1024, 1);
    gemm(GH, Wg2p, bg2, XC, NB, 1024, 1024, 128, 128, 256, 0);   // xc[:, 0:128]

    // protein branch
    conv1d_k<<<(NB * 384 + 255) / 256, 256, 0, stream>>>(tgt, Kxt, bxt, CONV);
    gemm(CONV, Wxtp, bxt2, XC + 128, NB, 384, 384, 128, 128, 256, 0);  // xc[:,128:256]

    // fused head
    gemm(XC, Wf1p, bf1, F1, NB, 256, 256, 1024, 1024, 1024, 1);
    gemm(F1, Wf2p, bf2, F2, NB, 1024, 1024, 512, 512, 512, 1);
    gemm(F2, Wop,  bo, (float*)d_out, NB, 512, 512, 16, 1, 1, 0);
}